// Linear_extractor_cluster_3126736192109
// MI455X (gfx1250) — compile-verified
//
#include <hip/hip_runtime.h>
#include <hip/hip_bf16.h>
#include <math.h>

// Problem constants
#define Bq 4096
#define Lq 512
#define Nq 8
#define Eq 8
#define Kq 2
#define Dq 256
#define Y_ELEMS (Bq * Dq * Nq)   // 8,388,608

typedef __attribute__((ext_vector_type(2))) float v2f;
typedef __attribute__((ext_vector_type(8))) float v8f;

// ---------------------------------------------------------------------------
// Kernel 0: zero y (atomically accumulated) and ws header (counts/imp/load)
// ---------------------------------------------------------------------------
__global__ __launch_bounds__(256) void zero_kernel(float* __restrict__ y,
                                                   float* __restrict__ ws_hdr) {
  unsigned i = blockIdx.x * 256u + threadIdx.x;
  float4 z = {0.f, 0.f, 0.f, 0.f};
  ((float4*)y)[i] = z;  // grid sized exactly: 8192*256*4 == Y_ELEMS
  if (blockIdx.x == 0 && threadIdx.x < 24) ws_hdr[threadIdx.x] = 0.f;  // cnt/imp/load
}

// ---------------------------------------------------------------------------
// Kernel 1: gating. One workgroup (256 thr = 8 waves) per token b.
//  - feats[l] = mean_n x[b,l,n]; logits = feats@Wg+bg ; noise path via Wn
//  - softmax, top-3, gates, importance/load atomics, expert dispatch lists
// ---------------------------------------------------------------------------
__global__ __launch_bounds__(256) void gating_kernel(
    const float* __restrict__ x, const float* __restrict__ noise,
    const float* __restrict__ Wg, const float* __restrict__ bg,
    const float* __restrict__ Wn, const float* __restrict__ bn,
    int* __restrict__ cnt, float* __restrict__ imp, float* __restrict__ loadv,
    int* __restrict__ toks, float* __restrict__ tgate) {
  const int b = blockIdx.x;
  const int tid = threadIdx.x;
  const float* xb = x + (size_t)b * Lq * Nq;

  float vals[16];
#pragma unroll
  for (int ch = 0; ch < 16; ++ch) vals[ch] = 0.f;

#pragma unroll
  for (int rep = 0; rep < 2; ++rep) {
    const int l = tid + rep * 256;
    const float4* xr = (const float4*)(xb + l * Nq);
    float4 a = xr[0], c = xr[1];
    float f = (a.x + a.y + a.z + a.w + c.x + c.y + c.z + c.w) * 0.125f;
    float wgl[8], wnl[8];
    ((float4*)wgl)[0] = ((const float4*)(Wg + l * Eq))[0];
    ((float4*)wgl)[1] = ((const float4*)(Wg + l * Eq))[1];
    ((float4*)wnl)[0] = ((const float4*)(Wn + l * Eq))[0];
    ((float4*)wnl)[1] = ((const float4*)(Wn + l * Eq))[1];
#pragma unroll
    for (int e = 0; e < 8; ++e) {
      vals[e] += f * wgl[e];
      vals[8 + e] += f * wnl[e];
    }
  }

  // wave32 shuffle reduction
#pragma unroll
  for (int ch = 0; ch < 16; ++ch) {
    float v = vals[ch];
#pragma unroll
    for (int off = 16; off > 0; off >>= 1) v += __shfl_down(v, off);
    vals[ch] = v;
  }

  __shared__ float sAcc[8][16];
  const int wave = tid >> 5, lane = tid & 31;
  if (lane == 0) {
#pragma unroll
    for (int ch = 0; ch < 16; ++ch) sAcc[wave][ch] = vals[ch];
  }
  __syncthreads();

  if (tid == 0) {
    float clean[8], stdv[8], noisy[8], probs[8];
#pragma unroll
    for (int e = 0; e < 8; ++e) {
      float c = bg[e], r = bn[e];
      for (int w = 0; w < 8; ++w) {
        c += sAcc[w][e];
        r += sAcc[w][8 + e];
      }
      clean[e] = c;
      float sp = (r > 20.f) ? r : log1pf(expf(r));  // softplus
      stdv[e] = sp + 0.01f;
      noisy[e] = c + noise[b * Eq + e] * stdv[e];
    }
    // softmax over E=8
    float m = noisy[0];
    for (int e = 1; e < 8; ++e) m = fmaxf(m, noisy[e]);
    float s = 0.f;
#pragma unroll
    for (int e = 0; e < 8; ++e) {
      probs[e] = expf(noisy[e] - m);
      s += probs[e];
    }
    float inv = 1.f / s;
#pragma unroll
    for (int e = 0; e < 8; ++e) probs[e] *= inv;

    // top-3 (selection; strict > matches lax.top_k tie-break by lowest index)
    float pc[8];
#pragma unroll
    for (int e = 0; e < 8; ++e) pc[e] = probs[e];
    int idx[3];
    float tv[3];
    for (int j = 0; j < 3; ++j) {
      int best = 0;
      float bv = pc[0];
      for (int e = 1; e < 8; ++e)
        if (pc[e] > bv) { bv = pc[e]; best = e; }
      idx[j] = best;
      tv[j] = bv;
      pc[best] = -1.0e30f;
    }
    const float tsum = tv[0] + tv[1] + 1e-6f;
    const float g0 = tv[0] / tsum, g1 = tv[1] / tsum;
    const float thr_in = tv[2], thr_out = tv[1];

    // load: Phi((clean - thr)/std), select by noisy > thr_in
#pragma unroll
    for (int e = 0; e < 8; ++e) {
      const float ss = fmaxf(stdv[e], 1e-9f);
      const float p_in = 0.5f * (1.f + erff((clean[e] - thr_in) / ss * 0.70710678f));
      const float p_out = 0.5f * (1.f + erff((clean[e] - thr_out) / ss * 0.70710678f));
      unsafeAtomicAdd(&loadv[e], (noisy[e] > thr_in) ? p_in : p_out);
    }
    // importance + dispatch
    unsafeAtomicAdd(&imp[idx[0]], g0);
    unsafeAtomicAdd(&imp[idx[1]], g1);
    int p0 = atomicAdd(&cnt[idx[0]], 1);
    toks[idx[0] * Bq + p0] = b;
    tgate[idx[0] * Bq + p0] = g0;
    int p1 = atomicAdd(&cnt[idx[1]], 1);
    toks[idx[1] * Bq + p1] = b;
    tgate[idx[1] * Bq + p1] = g1;
  }
}

// ---------------------------------------------------------------------------
// Kernel 2: grouped expert GEMM with WMMA fp32, double-buffered LDS pipeline.
// Block = (expert e, tile of 8 tokens). M=256 (all D), N=64 (8 tok x 8 n),
// K=512. 8 waves; wave w owns rows [32w,32w+32) => 2x4 grid of 16x16 C tiles.
// LDS strides 20 / 72 floats chosen for conflict-free WMMA fragment reads.
// Pipeline: stage chunk i+1's global loads into registers right after the
// barrier so vmem latency overlaps the 32 WMMAs of chunk i; one barrier/iter
// is sufficient with two buffers.
// ---------------------------------------------------------------------------
__global__ __launch_bounds__(256) void moe_gemm(
    const float* __restrict__ x, const float* __restrict__ We,
    const int* __restrict__ cnt, const int* __restrict__ toks,
    const float* __restrict__ tgate, float* __restrict__ y) {
  const int e = blockIdx.y;
  const int count = cnt[e];
  const int base = blockIdx.x * 8;
  if (base >= count) return;  // uniform exit; EXEC stays all-1s for WMMA
  const int nvalid = min(8, count - base);

  __shared__ int sb[8];
  __shared__ float sg[8];
  __shared__ float sWe[2][256 * 20];  // [d][k] stride 20
  __shared__ float sX[2][16 * 72];    // [k][col] stride 72, col = t*8+n

  const int tid = threadIdx.x;
  if (tid < 8) {
    const int ii = (tid < nvalid) ? (base + tid) : base;  // pad with valid token
    sb[tid] = toks[e * Bq + ii];
    sg[tid] = (tid < nvalid) ? tgate[e * Bq + ii] : 0.f;
  }
  __syncthreads();

  const float* WeE = We + (size_t)e * Dq * Lq;
  const int wave = tid >> 5, lane = tid & 31;
  const int h = lane >> 4, ln = lane & 15;
  const int mbase = wave * 32;

  const int tt = tid >> 4;  // token slot for x staging (tid<128)
  const int kk = tid & 15;
  const float* werow = WeE + (size_t)tid * Lq;
  const float* xrow =
      (tid < 128) ? (x + (size_t)sb[tt] * Lq * Nq + kk * Nq) : x;

  // register staging for the global->LDS pipeline
  float4 w0, w1, w2, w3, x0, x1;
  auto stage_global = [&](int l0) {
    const float4* wr = (const float4*)(werow + l0);
    w0 = wr[0]; w1 = wr[1]; w2 = wr[2]; w3 = wr[3];
    const float4* xr = (const float4*)(xrow + l0 * Nq);
    if (tid < 128) { x0 = xr[0]; x1 = xr[1]; }
    if (l0 + 16 < Lq) __builtin_prefetch(werow + l0 + 16);  // global_prefetch_b8
  };

  v8f acc[2][4] = {};
  stage_global(0);
  int buf = 0;

  for (int l0 = 0; l0 < Lq; l0 += 16) {
    // commit staged registers to LDS buffer `buf`
    float4* wd = (float4*)(&sWe[buf][tid * 20]);
    wd[0] = w0; wd[1] = w1; wd[2] = w2; wd[3] = w3;
    if (tid < 128) {
      float4* xd = (float4*)(&sX[buf][kk * 72 + tt * 8]);
      xd[0] = x0; xd[1] = x1;
    }
    __syncthreads();

    // issue next chunk's global loads now; they overlap the WMMAs below
    if (l0 + 16 < Lq) stage_global(l0 + 16);

#pragma unroll
    for (int k0 = 0; k0 < 16; k0 += 4) {
      v2f afr[2], bfr[4];
#pragma unroll
      for (int mt = 0; mt < 2; ++mt) {  // A 16x4: v0->K{0,2}, v1->K{1,3}
        const int m = mbase + mt * 16 + ln;
        afr[mt].x = sWe[buf][m * 20 + k0 + 2 * h];
        afr[mt].y = sWe[buf][m * 20 + k0 + 2 * h + 1];
      }
#pragma unroll
      for (int nt = 0; nt < 4; ++nt) {  // B 4x16: lanes0-15 K{0,1}, 16-31 K{2,3}
        const int col = nt * 16 + ln;
        bfr[nt].x = sX[buf][(k0 + 2 * h) * 72 + col];
        bfr[nt].y = sX[buf][(k0 + 2 * h + 1) * 72 + col];
      }
#pragma unroll
      for (int mt = 0; mt < 2; ++mt)
#pragma unroll
        for (int nt = 0; nt < 4; ++nt)
          acc[mt][nt] = __builtin_amdgcn_wmma_f32_16x16x4_f32(
              false, afr[mt], false, bfr[nt], (short)0, acc[mt][nt], false,
              false);
    }
    buf ^= 1;
  }

  // Epilogue: C layout VGPR r -> M = r + 8*(lane>=16), N = lane%16
#pragma unroll
  for (int nt = 0; nt < 4; ++nt) {
    const int col = nt * 16 + ln;
    if (col < nvalid * 8) {
      const int ts = col >> 3, nn = col & 7;
      const float g = sg[ts];
      float* yb = y + (size_t)sb[ts] * Dq * Nq + nn;
#pragma unroll
      for (int mt = 0; mt < 2; ++mt) {
#pragma unroll
        for (int r = 0; r < 8; ++r) {
          const int d = mbase + mt * 16 + 8 * h + r;
          unsafeAtomicAdd(yb + d * Nq, g * acc[mt][nt][r]);
        }
      }
    }
  }
}

// ---------------------------------------------------------------------------
// Kernel 3: losses + small outputs
// ---------------------------------------------------------------------------
__global__ void finalize_kernel(const int* __restrict__ cnt,
                                const float* __restrict__ imp,
                                const float* __restrict__ loadv,
                                float* __restrict__ tail) {
  if (threadIdx.x == 0 && blockIdx.x == 0) {
    float mi = 0.f, ml = 0.f;
    for (int e = 0; e < 8; ++e) { mi += imp[e]; ml += loadv[e]; }
    mi *= 0.125f; ml *= 0.125f;
    float vi = 0.f, vl = 0.f;
    for (int e = 0; e < 8; ++e) {
      const float a = imp[e] - mi, b = loadv[e] - ml;
      vi += a * a; vl += b * b;
    }
    vi *= (1.f / 7.f); vl *= (1.f / 7.f);  // ddof=1
    tail[0] = vi / (mi * mi + 1e-10f) + vl / (ml * ml + 1e-10f);
    for (int e = 0; e < 8; ++e) tail[1 + e] = imp[e] * (1.f / (float)Bq);
    for (int e = 0; e < 8; ++e) tail[9 + e] = (float)cnt[e];
  }
}

// ---------------------------------------------------------------------------
extern "C" void kernel_launch(void* const* d_in, const int* in_sizes, int n_in,
                              void* d_out, int out_size, void* d_ws,
                              size_t ws_size, hipStream_t stream) {
  const float* x = (const float*)d_in[0];
  const float* noise = (const float*)d_in[1];
  const float* Wg = (const float*)d_in[2];
  const float* bg = (const float*)d_in[3];
  const float* Wn = (const float*)d_in[4];
  const float* bn = (const float*)d_in[5];
  const float* We = (const float*)d_in[6];
  float* out = (float*)d_out;

  char* ws = (char*)d_ws;
  int* cnt = (int*)ws;                             // 8 ints
  float* imp = (float*)(ws + 32);                  // 8 floats
  float* loadv = (float*)(ws + 64);                // 8 floats
  int* toks = (int*)(ws + 128);                    // E*B ints
  float* tg = (float*)(ws + 128 + Eq * Bq * 4);    // E*B floats

  float* y = out;  // first Y_ELEMS floats of output

  zero_kernel<<<Y_ELEMS / (256 * 4), 256, 0, stream>>>(y, (float*)ws);
  gating_kernel<<<Bq, 256, 0, stream>>>(x, noise, Wg, bg, Wn, bn, cnt, imp,
                                        loadv, toks, tg);
  moe_gemm<<<dim3(Bq / 8, Eq), 256, 0, stream>>>(x, We, cnt, toks, tg, y);
  finalize_kernel<<<1, 32, 0, stream>>>(cnt, imp, loadv, out + Y_ELEMS);
}